// EquivariantLayerNormFast_16183436772004
// MI455X (gfx1250) — compile-verified
//
#include <hip/hip_runtime.h>

// EquivariantLayerNormFast for MI455X (gfx1250).
// Memory-bound: 768 MB moved, ~33 us floor at 23.3 TB/s.
// Strategy: wave32 per row, float4 (B128) non-temporal loads/stores,
// shfl_xor wave reductions, TDM tensor_load_to_lds for the affine tables.

typedef __attribute__((ext_vector_type(4))) float        v4f;
typedef __attribute__((ext_vector_type(4))) unsigned int u32x4;
typedef __attribute__((ext_vector_type(8))) int          i32x8;
typedef __attribute__((ext_vector_type(4))) int          i32x4;

#define FEAT 480
#define EPSF 1e-5f

__device__ __forceinline__ float wave_sum(float v) {
  v += __shfl_xor(v, 1, 32);
  v += __shfl_xor(v, 2, 32);
  v += __shfl_xor(v, 4, 32);
  v += __shfl_xor(v, 8, 32);
  v += __shfl_xor(v, 16, 32);
  return v;
}

#if __has_builtin(__builtin_amdgcn_tensor_load_to_lds) && __has_builtin(__builtin_amdgcn_s_wait_tensorcnt)
#define HAVE_TDM 1
#else
#define HAVE_TDM 0
#endif

#if HAVE_TDM
// Issue one TDM 1-D tile load: nelem f32 from gptr -> LDS byte address lds_addr.
// D# built per CDNA5 ISA ch.8 (group0: count/lds_addr/global_addr/type=2;
// group1: data_size=4B, tensor_dim0=tile_dim0=nelem, dim1=1, stride0=nelem).
__device__ __forceinline__ void tdm_load_1d(const void* gptr, unsigned lds_addr,
                                            unsigned nelem) {
  unsigned long long ga = (unsigned long long)gptr;
  u32x4 g0;
  g0.x = 1u;                                                // count=1 (valid)
  g0.y = lds_addr;                                          // lds_addr [63:32]
  g0.z = (unsigned)ga;                                      // global_addr lo
  g0.w = (unsigned)((ga >> 32) & 0x01FFFFFFu) | (2u << 30); // ga[56:32] | type=2
  i32x8 g1;
  g1[0] = (int)(2u << 16);                                  // data_size = 4 bytes
  g1[1] = (int)((nelem & 0xFFFFu) << 16);                   // tensor_dim0 lo16
  g1[2] = (int)(((nelem >> 16) & 0xFFFFu) | (1u << 16));    // dim0 hi16 | tensor_dim1=1
  g1[3] = (int)((nelem & 0xFFFFu) << 16);                   // tile_dim0 = nelem
  g1[4] = 1;                                                // tile_dim1 = 1
  g1[5] = (int)nelem;                                       // tensor_dim0_stride
  g1[6] = 0;
  g1[7] = 0;
  i32x4 z4 = (i32x4)0;
#if __clang_major__ >= 23
  i32x8 z8 = (i32x8)0;
  __builtin_amdgcn_tensor_load_to_lds(g0, g1, z4, z4, z8, 0);
#else
  __builtin_amdgcn_tensor_load_to_lds(g0, g1, z4, z4, 0);
#endif
}
#endif

__global__ __launch_bounds__(256) void eqln_kernel(
    const float* __restrict__ x, const float* __restrict__ w,
    const float* __restrict__ bias, float* __restrict__ out, int n) {
  __shared__ float smem[352];  // [0..223] = affine_weight, [224..351] = affine_bias

#if HAVE_TDM
  // Wave-relative LDS byte offset of smem: low 32 bits of the generic address
  // (flat aperture maps addr[31:0] directly to the logical LDS address).
  const unsigned lds_base = (unsigned)(unsigned long long)(uintptr_t)&smem[0];
  if ((threadIdx.x >> 5) == 0) {            // wave 0 only (uniform branch)
    tdm_load_1d(w, lds_base, 224u);
    tdm_load_1d(bias, lds_base + 224u * 4u, 128u);
    __builtin_amdgcn_s_wait_tensorcnt(0);
  }
  // The TDM builtin's LDS writes are invisible to the optimizer (descriptor
  // carries the LDS offset as an integer). Clobber memory so the smem loads
  // below are not folded to undef / dead-code-eliminated.
  asm volatile("" ::: "memory");
#else
  for (int i = threadIdx.x; i < 352; i += blockDim.x)
    smem[i] = (i < 224) ? w[i] : bias[i - 224];
#endif
  __syncthreads();

  const float* sw = smem;        // 224 weights
  const float* sb = smem + 224;  // 128 biases

  const int lane = threadIdx.x & 31;
  const long long wid = (long long)blockIdx.x * (blockDim.x >> 5) + (threadIdx.x >> 5);
  const long long nw  = (long long)gridDim.x * (blockDim.x >> 5);

  for (long long row = wid; row < n; row += nw) {
    const float* rp = x + (size_t)row * FEAT;
    float*       op = out + (size_t)row * FEAT;

    // 480 f32 = 4 float4 per lane; every float4 lies within one field
    // (field starts 0,128,320 are all multiples of 4).
    v4f a  = __builtin_nontemporal_load((const v4f*)rp + lane);          // 0..127   f1
    v4f bv = __builtin_nontemporal_load((const v4f*)(rp + 128) + lane);  // 128..255 f2
    v4f cv = __builtin_nontemporal_load((const v4f*)(rp + 256) + lane);  // 256..383 f2|f3
    v4f dv = (v4f)0.0f;
    if (lane < 24)
      dv = __builtin_nontemporal_load((const v4f*)(rp + 384) + lane);    // 384..479 f3

    float s1 = a.x + a.y + a.z + a.w;
    float q1 = a.x * a.x + a.y * a.y + a.z * a.z + a.w * a.w;
    float qb = bv.x * bv.x + bv.y * bv.y + bv.z * bv.z + bv.w * bv.w;
    float qc = cv.x * cv.x + cv.y * cv.y + cv.z * cv.z + cv.w * cv.w;
    float qd = dv.x * dv.x + dv.y * dv.y + dv.z * dv.z + dv.w * dv.w;

    float q2 = qb + ((lane < 16) ? qc : 0.0f);   // field2 squares (192 elems)
    float q3 = ((lane >= 16) ? qc : 0.0f) + qd;  // field3 squares (160 elems)

    s1 = wave_sum(s1);
    q1 = wave_sum(q1);
    q2 = wave_sum(q2);
    q3 = wave_sum(q3);

    const float mean = s1 * (1.0f / 128.0f);
    const float var  = q1 * (1.0f / 128.0f) - mean * mean;
    const float rstd = rsqrtf(var + EPSF);
    const float inv2 = rsqrtf(q2 * (1.0f / 192.0f) + EPSF);
    const float inv3 = rsqrtf(q3 * (1.0f / 160.0f) + EPSF);

    // ---- field 1: layernorm * w + b (w/b indices match element index) ----
    v4f o0;
    {
      const v4f wv = *((const v4f*)sw + lane);
      const v4f bb = *((const v4f*)sb + lane);
      o0.x = (a.x - mean) * rstd * wv.x + bb.x;
      o0.y = (a.y - mean) * rstd * wv.y + bb.y;
      o0.z = (a.z - mean) * rstd * wv.z + bb.z;
      o0.w = (a.w - mean) * rstd * wv.w + bb.w;
    }

    // ---- field 2 (bv): in-field offset t = lane*4+k, m = t/3 ----
    v4f o1;
    {
      const unsigned t = (unsigned)lane * 4u;
      o1.x = bv.x * inv2 * sw[128 + (t + 0u) / 3u];
      o1.y = bv.y * inv2 * sw[128 + (t + 1u) / 3u];
      o1.z = bv.z * inv2 * sw[128 + (t + 2u) / 3u];
      o1.w = bv.w * inv2 * sw[128 + (t + 3u) / 3u];
    }

    // ---- mixed vector cv: lanes 0..15 tail of field2, lanes 16..31 head of field3 ----
    v4f o2;
    if (lane < 16) {
      const unsigned t = 128u + (unsigned)lane * 4u;
      o2.x = cv.x * inv2 * sw[128 + (t + 0u) / 3u];
      o2.y = cv.y * inv2 * sw[128 + (t + 1u) / 3u];
      o2.z = cv.z * inv2 * sw[128 + (t + 2u) / 3u];
      o2.w = cv.w * inv2 * sw[128 + (t + 3u) / 3u];
    } else {
      const unsigned u = ((unsigned)lane - 16u) * 4u;
      o2.x = cv.x * inv3 * sw[192 + (u + 0u) / 5u];
      o2.y = cv.y * inv3 * sw[192 + (u + 1u) / 5u];
      o2.z = cv.z * inv3 * sw[192 + (u + 2u) / 5u];
      o2.w = cv.w * inv3 * sw[192 + (u + 3u) / 5u];
    }

    // ---- field 3 tail (dv), lanes 0..23: u = 64 + lane*4 + k, m = u/5 ----
    if (lane < 24) {
      const unsigned u = 64u + (unsigned)lane * 4u;
      v4f o3;
      o3.x = dv.x * inv3 * sw[192 + (u + 0u) / 5u];
      o3.y = dv.y * inv3 * sw[192 + (u + 1u) / 5u];
      o3.z = dv.z * inv3 * sw[192 + (u + 2u) / 5u];
      o3.w = dv.w * inv3 * sw[192 + (u + 3u) / 5u];
      __builtin_nontemporal_store(o3, (v4f*)(op + 384) + lane);
    }
    __builtin_nontemporal_store(o0, (v4f*)op + lane);
    __builtin_nontemporal_store(o1, (v4f*)(op + 128) + lane);
    __builtin_nontemporal_store(o2, (v4f*)(op + 256) + lane);
  }
}

extern "C" void kernel_launch(void* const* d_in, const int* in_sizes, int n_in,
                              void* d_out, int out_size, void* d_ws, size_t ws_size,
                              hipStream_t stream) {
  const float* x    = (const float*)d_in[0];  // (N, 480) f32
  const float* w    = (const float*)d_in[1];  // (224,)   f32
  const float* bias = (const float*)d_in[2];  // (128,)   f32
  float* out = (float*)d_out;
  const int n = in_sizes[0] / FEAT;  // number of nodes

  const int block = 256;  // 8 wave32 per block
  int grid = 2048;        // grid-stride over rows, one wave per row
  if (grid * (block / 32) > n) grid = (n + (block / 32) - 1) / (block / 32);
  if (grid < 1) grid = 1;

  eqln_kernel<<<grid, block, 0, stream>>>(x, w, bias, out, n);
}